// InEKFLayer_35141422416177
// MI455X (gfx1250) — compile-verified
//
#include <hip/hip_runtime.h>
#include <math.h>

#define LIE 32
#define OBS 64
#define DD  64
#define TT  2048
#define BB  8
#define BT  (BB*TT)

typedef float v2f __attribute__((ext_vector_type(2)));
typedef float v8f __attribute__((ext_vector_type(8)));

// ---------------------------------------------------------------------------
// CDNA5 async cache<->LDS copies (ASYNCcnt-tracked), guarded so the file
// still compiles on toolchains without the builtins.
// Builtin pointer params are int4-vector pointers in AS1 (global) / AS3 (LDS),
// per the compiler diagnostics.
// ---------------------------------------------------------------------------
#if __has_builtin(__builtin_amdgcn_global_load_async_to_lds_b128) && \
    __has_builtin(__builtin_amdgcn_global_store_async_from_lds_b128) && \
    __has_builtin(__builtin_amdgcn_s_wait_asynccnt)
#define HAS_ASYNC_LDS 1
#else
#define HAS_ASYNC_LDS 0
#endif

typedef int av4i __attribute__((vector_size(16)));
typedef __attribute__((address_space(1))) av4i* g128_t;   // global int4*
typedef __attribute__((address_space(3))) av4i* l128_t;   // LDS int4*

__device__ __forceinline__ void async_ld_b128(void* lds, const void* g)
{
#if HAS_ASYNC_LDS
    __builtin_amdgcn_global_load_async_to_lds_b128((g128_t)g, (l128_t)lds, 0, 0);
#else
    *(float4*)lds = *(const float4*)g;
#endif
}

__device__ __forceinline__ void async_st_b128(void* g, const void* lds)
{
#if HAS_ASYNC_LDS
    __builtin_amdgcn_global_store_async_from_lds_b128((g128_t)g, (l128_t)lds, 0, 0);
#else
    *(float4*)g = *(const float4*)lds;
#endif
}

__device__ __forceinline__ void async_wait()
{
#if HAS_ASYNC_LDS
    __builtin_amdgcn_s_wait_asynccnt(0);
#endif
}

// ---------------------------------------------------------------------------
// 16x16 f32 WMMA tile helpers (V_WMMA_F32_16X16X4_F32, wave32).
// A 16x4 layout: lanes 0-15 -> M=lane, VGPR0/1 = K,K+1 ; lanes 16-31 -> K+2,K+3
// C/D 16x16 layout: VGPR v -> row v (lanes 0-15) / row v+8 (lanes 16-31)
// ---------------------------------------------------------------------------
__device__ __forceinline__ v8f wmma16x16_f32(const float* A, int lda, bool tA,
                                             const float* B, int ldb, bool tB,
                                             int K, v8f acc)
{
    const int lane = threadIdx.x & 31;
    const int half = lane >> 4;   // 0 or 1
    const int idx  = lane & 15;   // A row / B col within tile
    for (int k0 = 0; k0 < K; k0 += 4) {
        const int ka = k0 + 2 * half;
        v2f a, b;
        if (!tA) { a[0] = A[idx * lda + ka];     a[1] = A[idx * lda + ka + 1]; }
        else     { a[0] = A[ka * lda + idx];     a[1] = A[(ka + 1) * lda + idx]; }
        if (!tB) { b[0] = B[ka * ldb + idx];     b[1] = B[(ka + 1) * ldb + idx]; }
        else     { b[0] = B[idx * ldb + ka];     b[1] = B[idx * ldb + ka + 1]; }
        acc = __builtin_amdgcn_wmma_f32_16x16x4_f32(false, a, false, b,
                                                    (short)0, acc, false, false);
    }
    return acc;
}

__device__ __forceinline__ v8f wmma_load_c(const float* C, int ldc)
{
    const int lane = threadIdx.x & 31;
    const int half = lane >> 4;
    const int idx  = lane & 15;
    v8f acc;
#pragma unroll
    for (int v = 0; v < 8; ++v) acc[v] = C[(v + 8 * half) * ldc + idx];
    return acc;
}

__device__ __forceinline__ void wmma_store_c(float* C, int ldc, v8f acc)
{
    const int lane = threadIdx.x & 31;
    const int half = lane >> 4;
    const int idx  = lane & 15;
#pragma unroll
    for (int v = 0; v < 8; ++v) C[(v + 8 * half) * ldc + idx] = acc[v];
}

// C[MxN] (row-major, ldc) = op(A)*op(B) (+C if accum). M,N multiples of 16.
// `wave` is an SGPR (readfirstlane'd by the caller) so the tile loop uses
// scalar branches and EXEC stays all-1s across every WMMA (ISA requirement).
__device__ __forceinline__ void gemm_lds(float* C, int ldc,
                                         const float* A, int lda, bool tA,
                                         const float* B, int ldb, bool tB,
                                         int M, int N, int K,
                                         int wave, int nwaves, bool accum)
{
    const int tm = M >> 4, tn = N >> 4;
    for (int t = wave; t < tm * tn; t += nwaves) {
        const int ti = t / tn, tj = t % tn;
        const float* Ap = tA ? (A + ti * 16) : (A + ti * 16 * lda);
        const float* Bp = tB ? (B + tj * 16 * ldb) : (B + tj * 16);
        float* Cp = C + ti * 16 * ldc + tj * 16;
        v8f acc;
        if (accum) acc = wmma_load_c(Cp, ldc);
        else { v8f z = {}; acc = z; }
        acc = wmma16x16_f32(Ap, lda, tA, Bp, ldb, tB, K, acc);
        wmma_store_c(Cp, ldc, acc);
    }
}

__device__ __forceinline__ int uniform_wave_id()
{
    return __builtin_amdgcn_readfirstlane((int)(threadIdx.x >> 5));
}

// ---------------------------------------------------------------------------
// Phase 1: z = ce @ obs_w^T + obs_b ; h = relu(ce @ det_w1^T + b1)
//          probs = sigmoid(h @ det_w2^T + b2)
// One block handles 64 rows of the flattened (B*T, 64) activation matrix.
// ---------------------------------------------------------------------------
__global__ __launch_bounds__(256) void ekf_precompute(
    const float* __restrict__ ce,
    const float* __restrict__ det_w1, const float* __restrict__ det_b1,
    const float* __restrict__ det_w2, const float* __restrict__ det_b2,
    const float* __restrict__ obs_w,  const float* __restrict__ obs_b,
    float* __restrict__ z_ws, float* __restrict__ probs_out)
{
    __shared__ float sCE[64 * 64];
    __shared__ float sW[64 * 64];
    __shared__ float sO[64 * 64];

    const int tid  = threadIdx.x;
    const int wave = uniform_wave_id();
    const int nw   = blockDim.x >> 5;
    const int row0 = blockIdx.x * 64;

    // 16 KB tile + 16 KB weights via ASYNCcnt-tracked cache->LDS DMA
    for (int i = 0; i < 4; ++i) {
        const int o = i * 1024 + tid * 4;          // floats
        async_ld_b128(&sCE[o], &ce[(size_t)row0 * 64 + o]);
        async_ld_b128(&sW[o], &obs_w[o]);
    }
    async_wait();
    __syncthreads();

    // z = ce * obs_w^T
    gemm_lds(sO, 64, sCE, 64, false, sW, 64, true, 64, 64, 64, wave, nw, false);
    __syncthreads();
    for (int i = tid; i < 64 * 64; i += blockDim.x)
        z_ws[(size_t)row0 * 64 + i] = sO[i] + obs_b[i & 63];
    __syncthreads();

    for (int i = 0; i < 4; ++i) {
        const int o = i * 1024 + tid * 4;
        async_ld_b128(&sW[o], &det_w1[o]);
    }
    async_wait();
    __syncthreads();

    // h = relu(ce * det_w1^T + b1)
    gemm_lds(sO, 64, sCE, 64, false, sW, 64, true, 64, 64, 64, wave, nw, false);
    __syncthreads();
    for (int i = tid; i < 64 * 64; i += blockDim.x)
        sO[i] = fmaxf(sO[i] + det_b1[i & 63], 0.0f);
    __syncthreads();

    if (tid < 64) {
        float acc = det_b2[0];
        for (int c = 0; c < 64; ++c) acc += sO[tid * 64 + c] * det_w2[c];
        probs_out[row0 + tid] = 1.0f / (1.0f + __expf(-acc));
    }
}

// ---------------------------------------------------------------------------
// Phase 2: sequential Kalman covariance scan. One workgroup per batch chain.
// Sigma lives in LDS across all T steps; all GEMMs via WMMA; 64x64 SPD
// inverse via in-place Gauss-Jordan in LDS.
// ---------------------------------------------------------------------------
__global__ __launch_bounds__(256) void ekf_scan(
    const float* __restrict__ Fg,   const float* __restrict__ logQ,
    const float* __restrict__ logR, const float* __restrict__ Hg,
    const float* __restrict__ probs,
    float* __restrict__ covs_out, float* __restrict__ K_ws)
{
    __shared__ float sF[LIE * LIE];     // 4 KB
    __shared__ float sH[OBS * LIE];     // 8 KB
    __shared__ float Sig[LIE * LIE];    // 4 KB  (Sigma / Sigma_u)
    __shared__ float P[LIE * LIE];      // 4 KB  (predicted)
    __shared__ float T1[LIE * LIE];     // 4 KB  (temp)
    __shared__ float HS[OBS * LIE];     // 8 KB  (H*P, later K*diag(R))
    __shared__ float Sm[OBS * OBS];     // 16 KB (S -> S^-1 in place)
    __shared__ float Km[LIE * OBS];     // 8 KB
    __shared__ float IKH[LIE * LIE];    // 4 KB
    __shared__ float Qd[LIE];
    __shared__ float Rd[OBS];
    __shared__ float gp[BB];

    const int b    = blockIdx.x;
    const int tid  = threadIdx.x;
    const int wave = uniform_wave_id();
    const int nw   = blockDim.x >> 5;

    for (int i = tid; i < LIE * LIE; i += blockDim.x) {
        sF[i]  = Fg[i];
        Sig[i] = ((i / LIE) == (i % LIE)) ? 0.01f : 0.0f;
    }
    for (int i = tid; i < OBS * LIE; i += blockDim.x) sH[i] = Hg[i];
    if (tid < LIE) Qd[tid] = __expf(logQ[tid]);
    if (tid < OBS) Rd[tid] = __expf(logR[tid]);
    __syncthreads();

    for (int t = 0; t < TT; ++t) {
        if (tid < BB) gp[tid] = probs[tid * TT + t];

        // ---- predict: P = F * Sig * F^T + diag(Q)
        gemm_lds(T1, LIE, Sig, LIE, false, sF, LIE, true, LIE, LIE, LIE, wave, nw, false);
        __syncthreads();
        gemm_lds(P, LIE, sF, LIE, false, T1, LIE, false, LIE, LIE, LIE, wave, nw, false);
        __syncthreads();
        if (tid < LIE) P[tid * LIE + tid] += Qd[tid];
        __syncthreads();

        // ---- innovation covariance: S = H * P * H^T + diag(R)
        gemm_lds(HS, LIE, sH, LIE, false, P, LIE, false, OBS, LIE, LIE, wave, nw, false);
        __syncthreads();
        gemm_lds(Sm, OBS, HS, LIE, false, sH, LIE, true, OBS, OBS, LIE, wave, nw, false);
        __syncthreads();
        if (tid < OBS) Sm[tid * OBS + tid] += Rd[tid];
        __syncthreads();

        // ---- in-place Gauss-Jordan inverse of S (SPD, no pivoting)
        for (int p = 0; p < OBS; ++p) {
            const float ip = 1.0f / Sm[p * OBS + p];
            for (int c = tid; c < OBS; c += blockDim.x)
                if (c != p) Sm[p * OBS + c] *= ip;
            __syncthreads();
            const int r = tid & 63;
            const int q = tid >> 6;            // column quarter 0..3
            const float f = Sm[r * OBS + p];
            __syncthreads();
            if (r != p) {
                const int c0 = q * 16;
                for (int c = c0; c < c0 + 16; ++c)
                    if (c != p) Sm[r * OBS + c] -= f * Sm[p * OBS + c];
                if (q == (p >> 4)) Sm[r * OBS + p] = -f * ip;
            }
            if (tid == 0) Sm[p * OBS + p] = ip;
            __syncthreads();
        }

        // ---- gain: K = (H*P)^T * S^-1  (P symmetric => P H^T = HS^T)
        gemm_lds(Km, OBS, HS, LIE, true, Sm, OBS, false, LIE, OBS, OBS, wave, nw, false);
        __syncthreads();

        // ---- IKH = I - K*H
        gemm_lds(IKH, LIE, Km, OBS, false, sH, LIE, false, LIE, LIE, OBS, wave, nw, false);
        __syncthreads();
        for (int i = tid; i < LIE * LIE; i += blockDim.x) {
            const float d = ((i / LIE) == (i % LIE)) ? 1.0f : 0.0f;
            IKH[i] = d - IKH[i];
        }
        __syncthreads();

        // ---- Joseph: Sig_u = IKH*P*IKH^T + K*diag(R)*K^T
        gemm_lds(T1, LIE, IKH, LIE, false, P, LIE, false, LIE, LIE, LIE, wave, nw, false);
        __syncthreads();
        gemm_lds(Sig, LIE, T1, LIE, false, IKH, LIE, true, LIE, LIE, LIE, wave, nw, false);
        __syncthreads();
        for (int i = tid; i < LIE * OBS; i += blockDim.x)
            HS[i] = Km[i] * Rd[i & 63];          // reuse HS as K*diag(R)
        __syncthreads();
        gemm_lds(Sig, LIE, HS, OBS, false, Km, OBS, true, LIE, LIE, OBS, wave, nw, true);
        __syncthreads();

        // ---- blend, persist Sigma, emit covs and K
        bool gate = false;
        for (int i = 0; i < BB; ++i) gate = gate || (gp[i] > 0.5f);
        const float a = gp[b];
        for (int i = tid; i < LIE * LIE; i += blockDim.x) {
            const float so = gate ? (a * Sig[i] + (1.0f - a) * P[i]) : P[i];
            Sig[i] = so;
            covs_out[((size_t)b * TT + t) * (LIE * LIE) + i] = so;
        }
        for (int i = tid; i < LIE * OBS; i += blockDim.x)
            K_ws[((size_t)b * TT + t) * (LIE * OBS) + i] = Km[i];
        __syncthreads();
    }
}

// ---------------------------------------------------------------------------
// Phase 3: per-(b,t) correction of the 64x64 positions tile (bandwidth bound).
// Bulk traffic staged cache<->LDS with async B128 DMA.
// ---------------------------------------------------------------------------
__global__ __launch_bounds__(256) void ekf_correct(
    const float* __restrict__ pos, const float* __restrict__ Hg,
    const float* __restrict__ z_ws, const float* __restrict__ K_ws,
    const float* __restrict__ probs, float* __restrict__ corr_out)
{
    __shared__ float sP[DD * DD];       // 16 KB
    __shared__ float sK[LIE * OBS];     // 8 KB
    __shared__ float sz[OBS];
    __shared__ float sy[OBS];
    __shared__ float mu[LIE];
    __shared__ float muc[LIE];
    __shared__ float gp[BB];

    const int idx = blockIdx.x;         // b*T + t
    const int b   = idx / TT;
    const int t   = idx - b * TT;
    const int tid = threadIdx.x;
    const size_t base = (size_t)idx * DD * DD;

    // positions tile: 16 KB = 4 x (256 lanes x 16 B)
    for (int i = 0; i < 4; ++i) {
        const int o = i * 1024 + tid * 4;
        async_ld_b128(&sP[o], &pos[base + o]);
    }
    // gain tile: 8 KB = 2 x (256 lanes x 16 B)
    for (int i = 0; i < 2; ++i) {
        const int o = i * 1024 + tid * 4;
        async_ld_b128(&sK[o], &K_ws[(size_t)idx * (LIE * OBS) + o]);
    }
    if (tid < 16)                        // z row: 256 B
        async_ld_b128(&sz[tid * 4], &z_ws[(size_t)idx * OBS + tid * 4]);
    if (tid < BB)  gp[tid] = probs[tid * TT + t];
    if (idx + 1 < BT)                    // pull next tile toward L2/WGP$
        __builtin_prefetch(&pos[base + (size_t)DD * DD + (size_t)tid * 16], 0, 0);
    async_wait();
    __syncthreads();

    if (tid < LIE)
        mu[tid] = atan2f(sP[(2 * tid + 1) * DD + 2 * tid], sP[2 * tid * DD + 2 * tid]);
    __syncthreads();

    if (tid < OBS) {                     // y = z - H*mu
        float acc = sz[tid];
        for (int l = 0; l < LIE; ++l) acc -= mu[l] * Hg[tid * LIE + l];
        sy[tid] = acc;
    }
    __syncthreads();

    if (tid < LIE) {                     // mu_c = mu + K*y
        float acc = mu[tid];
        for (int o = 0; o < OBS; ++o) acc += sK[tid * OBS + o] * sy[o];
        muc[tid] = acc;
    }
    __syncthreads();

    bool gate = false;
    for (int i = 0; i < BB; ++i) gate = gate || (gp[i] > 0.5f);
    const float a = gp[b];

    if (gate && tid < LIE) {
        const float c = __cosf(muc[tid]);
        const float s = __sinf(muc[tid]);
        const int r0 = 2 * tid, c0 = 2 * tid;
        const float b00 = sP[r0 * DD + c0],       b01 = sP[r0 * DD + c0 + 1];
        const float b10 = sP[(r0 + 1) * DD + c0], b11 = sP[(r0 + 1) * DD + c0 + 1];
        sP[r0 * DD + c0]           = a * c    + (1.0f - a) * b00;
        sP[r0 * DD + c0 + 1]       = a * (-s) + (1.0f - a) * b01;
        sP[(r0 + 1) * DD + c0]     = a * s    + (1.0f - a) * b10;
        sP[(r0 + 1) * DD + c0 + 1] = a * c    + (1.0f - a) * b11;
    }
    __syncthreads();

    // stream the corrected tile out: LDS -> cache, ASYNCcnt-tracked
    for (int i = 0; i < 4; ++i) {
        const int o = i * 1024 + tid * 4;
        async_st_b128(&corr_out[base + o], &sP[o]);
    }
    async_wait();
}

// ---------------------------------------------------------------------------
extern "C" void kernel_launch(void* const* d_in, const int* in_sizes, int n_in,
                              void* d_out, int out_size, void* d_ws, size_t ws_size,
                              hipStream_t stream)
{
    const float* positions = (const float*)d_in[0];
    const float* content   = (const float*)d_in[1];
    const float* F         = (const float*)d_in[2];
    const float* logQ      = (const float*)d_in[3];
    const float* logR      = (const float*)d_in[4];
    const float* Hw        = (const float*)d_in[5];
    const float* det_w1    = (const float*)d_in[6];
    const float* det_b1    = (const float*)d_in[7];
    const float* det_w2    = (const float*)d_in[8];
    const float* det_b2    = (const float*)d_in[9];
    const float* obs_w     = (const float*)d_in[10];
    const float* obs_b     = (const float*)d_in[11];

    float* corr  = (float*)d_out;                      // (B,T,64,64)
    float* covs  = corr + (size_t)BT * DD * DD;        // (B,T,32,32)
    float* probs = covs + (size_t)BT * LIE * LIE;      // (B,T)

    float* z_ws = (float*)d_ws;                        // (B*T,64)
    float* K_ws = z_ws + (size_t)BT * OBS;             // (B*T,32,64)

    ekf_precompute<<<BT / 64, 256, 0, stream>>>(content, det_w1, det_b1,
                                                det_w2, det_b2, obs_w, obs_b,
                                                z_ws, probs);
    ekf_scan<<<BB, 256, 0, stream>>>(F, logQ, logR, Hw, probs, covs, K_ws);
    ekf_correct<<<BT, 256, 0, stream>>>(positions, Hw, z_ws, K_ws, probs, corr);
}